// GCN_7765300871431
// MI455X (gfx1250) — compile-verified
//
#include <hip/hip_runtime.h>

typedef __attribute__((ext_vector_type(2))) float v2f;
typedef __attribute__((ext_vector_type(8))) float v8f;

// ---------------- utility kernels ----------------

__global__ void zero_kernel(float* __restrict__ p, long long n) {
  long long i = (long long)blockIdx.x * blockDim.x + threadIdx.x;
  long long stride = (long long)gridDim.x * blockDim.x;
  for (; i < n; i += stride) p[i] = 0.0f;
}

__global__ void count_kernel(const int* __restrict__ idx, float* __restrict__ deg, int n) {
  int i = blockIdx.x * blockDim.x + threadIdx.x;
  if (i < n) atomicAdd(&deg[idx[i]], 1.0f);
}

__global__ void rsqrt_kernel(float* __restrict__ deg, long long n) {
  long long i = (long long)blockIdx.x * blockDim.x + threadIdx.x;
  long long stride = (long long)gridDim.x * blockDim.x;
  for (; i < n; i += stride) deg[i] = 1.0f / sqrtf(fmaxf(deg[i], 1.0f));
}

// ---------------- edge scatter: accum[dst] += x[src] * rs_src[src] ----------------
// One wave per edge; lane l carries floats [4l, 4l+4) of the 128-dim feature.
__global__ void aggregate_kernel(const float* __restrict__ x,
                                 const int* __restrict__ src,
                                 const int* __restrict__ dst,
                                 const float* __restrict__ rs_src,
                                 float* __restrict__ accum,
                                 int nedges) {
  int gw = (blockIdx.x * blockDim.x + threadIdx.x) >> 5;
  int lane = threadIdx.x & 31;
  int nw = (gridDim.x * blockDim.x) >> 5;
  for (int e = gw; e < nedges; e += nw) {
    int s = src[e];
    int d = dst[e];
    float sc = rs_src[s];
    float4 v = ((const float4*)(x + (long long)s * 128))[lane];
    float* o = accum + (long long)d * 128 + lane * 4;
    atomicAdd(o + 0, v.x * sc);
    atomicAdd(o + 1, v.y * sc);
    atomicAdd(o + 2, v.z * sc);
    atomicAdd(o + 3, v.w * sc);
  }
}

// ---------------- in-place fused GEMM: m <- relu((m * rs[:,None]) @ W + b) ----------------
// Block = 256 threads = 8 waves. Block handles rows [16*bid, 16*bid+16), all 128 cols.
// Wave w computes the 16x16 tile at cols [16w, 16w+16) with V_WMMA_F32_16X16X4_F32, K=128.
// A fragment layout (f32 16x4): lanes 0-15 -> M=lane, VGPR0/1 hold K={0,1}; lanes 16-31 hold K={2,3}.
// B fragment (4x16, column-major mirror): lane%16 -> N, lane/16 selects K pair.
// D layout: acc[j], M = j + 8*(lane/16), N = lane%16.
__global__ __launch_bounds__(256) void gemm_kernel(float* __restrict__ m,
                                                   const float* __restrict__ W,
                                                   const float* __restrict__ bias,
                                                   const float* __restrict__ rs) {
  __shared__ float As[16][132];  // pad 128 -> 132 floats: conflict-free column reads
  const int rowBase = blockIdx.x * 16;
  const int tid = threadIdx.x;
  // cooperative stage of the 16x128 A block into LDS (float4 granularity)
  for (int i = tid; i < 16 * 32; i += 256) {
    int r = i >> 5, c4 = i & 31;
    float4 v = ((const float4*)(m + (long long)(rowBase + r) * 128))[c4];
    *(float4*)&As[r][c4 * 4] = v;
  }
  __syncthreads();

  const int wave = tid >> 5;
  const int lane = tid & 31;
  const int colBase = wave * 16;
  const int l16 = lane & 15;    // M for A-frag, N for B-frag / D
  const int khalf = lane >> 4;  // selects K pair {0,1} vs {2,3}

  v8f acc = {};
  for (int k0 = 0; k0 < 128; k0 += 4) {
    const int k = k0 + 2 * khalf;
    v2f a, b;
    a.x = As[l16][k + 0];
    a.y = As[l16][k + 1];
    const float* Wp = W + (long long)k * 128 + colBase + l16;
    b.x = Wp[0];
    b.y = Wp[128];
    acc = __builtin_amdgcn_wmma_f32_16x16x4_f32(
        /*neg_a=*/false, a, /*neg_b=*/false, b,
        /*c_mod=*/(short)0, acc, /*reuse_a=*/false, /*reuse_b=*/false);
  }

  const float bc = bias[colBase + l16];
#pragma unroll
  for (int j = 0; j < 8; ++j) {
    const int row = rowBase + j + 8 * khalf;
    // row scale commutes with the matmul: (rs*m) @ W == rs * (m @ W)
    float v = fmaf(acc[j], rs[row], bc);
    m[(long long)row * 128 + colBase + l16] = fmaxf(v, 0.0f);
  }
}

// ---------------- host orchestration ----------------

extern "C" void kernel_launch(void* const* d_in, const int* in_sizes, int n_in,
                              void* d_out, int out_size, void* d_ws, size_t ws_size,
                              hipStream_t stream) {
  (void)n_in; (void)out_size; (void)ws_size;

  const float* h_p   = (const float*)d_in[0];
  const float* h_d   = (const float*)d_in[1];
  const int*   pd_src = (const int*)d_in[2];
  const int*   pd_dst = (const int*)d_in[3];
  const int*   dp_src = (const int*)d_in[4];
  const int*   dp_dst = (const int*)d_in[5];
  const float* W1_pd = (const float*)d_in[6];
  const float* b1_pd = (const float*)d_in[7];
  const float* W1_dp = (const float*)d_in[8];
  const float* b1_dp = (const float*)d_in[9];
  const float* W2_pd = (const float*)d_in[10];
  const float* b2_pd = (const float*)d_in[11];
  const float* W2_dp = (const float*)d_in[12];
  const float* b2_dp = (const float*)d_in[13];
  const float* W3_pd = (const float*)d_in[14];
  const float* b3_pd = (const float*)d_in[15];
  const float* W3_dp = (const float*)d_in[16];
  const float* b3_dp = (const float*)d_in[17];

  const int NP = in_sizes[0] / 128;
  const int ND = in_sizes[1] / 128;
  const int E  = in_sizes[2];

  // workspace layout (floats): Bp[NP*128] | Bd1[ND*128] | Bd2[ND*128] | deg/rs vectors
  float* Bp  = (float*)d_ws;
  float* Bd1 = Bp  + (long long)NP * 128;
  float* Bd2 = Bd1 + (long long)ND * 128;
  float* rs_p_out = Bd2 + (long long)ND * 128;  // deg of p as pd-src
  float* rs_d_in  = rs_p_out + NP;              // deg of d as pd-dst
  float* rs_d_out = rs_d_in + ND;               // deg of d as dp-src
  float* rs_p_in  = rs_d_out + ND;              // deg of p as dp-dst

  float* out_p = (float*)d_out;                 // h_p3 [NP,128]
  float* out_d = out_p + (long long)NP * 128;   // h_d3 [ND,128]

  const int ZB = 2048;
  const int CT = 256;
  const int AGG_BLOCKS = 4096, AGG_T = 256;     // 32768 waves, grid-stride over edges
  auto cdiv = [](int a, int b) { return (a + b - 1) / b; };

  // ---- degrees (shared by all 6 gconvs) ----
  const long long degTotal = 2LL * NP + 2LL * ND;  // contiguous region
  zero_kernel<<<ZB, 256, 0, stream>>>(rs_p_out, degTotal);
  count_kernel<<<cdiv(E, CT), CT, 0, stream>>>(pd_src, rs_p_out, E);
  count_kernel<<<cdiv(E, CT), CT, 0, stream>>>(pd_dst, rs_d_in, E);
  count_kernel<<<cdiv(E, CT), CT, 0, stream>>>(dp_src, rs_d_out, E);
  count_kernel<<<cdiv(E, CT), CT, 0, stream>>>(dp_dst, rs_p_in, E);
  rsqrt_kernel<<<ZB, 256, 0, stream>>>(rs_p_out, degTotal);

  // ---- layer 1 ----
  zero_kernel<<<ZB, 256, 0, stream>>>(Bd1, (long long)ND * 128);
  aggregate_kernel<<<AGG_BLOCKS, AGG_T, 0, stream>>>(h_p, pd_src, pd_dst, rs_p_out, Bd1, E);
  gemm_kernel<<<ND / 16, 256, 0, stream>>>(Bd1, W1_pd, b1_pd, rs_d_in);   // Bd1 = h_d1

  zero_kernel<<<ZB, 256, 0, stream>>>(Bp, (long long)NP * 128);
  aggregate_kernel<<<AGG_BLOCKS, AGG_T, 0, stream>>>(h_d, dp_src, dp_dst, rs_d_out, Bp, E);
  gemm_kernel<<<NP / 16, 256, 0, stream>>>(Bp, W1_dp, b1_dp, rs_p_in);    // Bp = h_p1

  // ---- layer 2 ----
  zero_kernel<<<ZB, 256, 0, stream>>>(Bd2, (long long)ND * 128);
  aggregate_kernel<<<AGG_BLOCKS, AGG_T, 0, stream>>>(Bp, pd_src, pd_dst, rs_p_out, Bd2, E);
  gemm_kernel<<<ND / 16, 256, 0, stream>>>(Bd2, W2_pd, b2_pd, rs_d_in);   // Bd2 = h_d2

  zero_kernel<<<ZB, 256, 0, stream>>>(Bp, (long long)NP * 128);           // h_p1 dead
  aggregate_kernel<<<AGG_BLOCKS, AGG_T, 0, stream>>>(Bd1, dp_src, dp_dst, rs_d_out, Bp, E);
  gemm_kernel<<<NP / 16, 256, 0, stream>>>(Bp, W2_dp, b2_dp, rs_p_in);    // Bp = h_p2

  // ---- layer 3 (write straight into d_out) ----
  zero_kernel<<<ZB, 256, 0, stream>>>(out_d, (long long)ND * 128);
  aggregate_kernel<<<AGG_BLOCKS, AGG_T, 0, stream>>>(Bp, pd_src, pd_dst, rs_p_out, out_d, E);
  gemm_kernel<<<ND / 16, 256, 0, stream>>>(out_d, W3_pd, b3_pd, rs_d_in); // h_d3

  zero_kernel<<<ZB, 256, 0, stream>>>(out_p, (long long)NP * 128);
  aggregate_kernel<<<AGG_BLOCKS, AGG_T, 0, stream>>>(Bd2, dp_src, dp_dst, rs_d_out, out_p, E);
  gemm_kernel<<<NP / 16, 256, 0, stream>>>(out_p, W3_dp, b3_dp, rs_p_in); // h_p3
}